// BERT_CRF_44298292691565
// MI455X (gfx1250) — compile-verified
//
#include <hip/hip_runtime.h>
#include <hip/hip_bf16.h>

// ---------------- problem constants (from reference) ----------------
#define BDIM 64
#define TDIM 512
#define HDIM 768
#define KDIM 6
#define KPAD 16          // B/N dimension padded to WMMA width
#define START_IDX 4
#define STOP_IDX 5
#define NEGV -10000.0f

typedef float v2f __attribute__((ext_vector_type(2)));
typedef float v8f __attribute__((ext_vector_type(8)));

// =====================================================================
// Kernel 1: feats[r, k] = sum_h x[r, h] * W[h, k] + b[k]
// r = flattened (batch, t) row, 32768 rows. One wave per 16-row tile.
// V_WMMA_F32_16X16X4_F32: A = 16x4 tile of x, B = 4x16 of W.
// W is staged in LDS PADDED to 16 columns (zeros for k>=6) so the
// inner-loop B load is unconditional -> no EXEC save/restore, no
// zero-fill movs in the hot loop. Bandwidth-bound on x (one pass).
// =====================================================================
__global__ __launch_bounds__(256) void
bert_crf_gemm_feats(const float* __restrict__ x,
                    const float* __restrict__ W,
                    const float* __restrict__ bias,
                    float* __restrict__ feats,
                    int nrows) {
    __shared__ float sW[HDIM * KPAD];          // 48 KB, zero-padded columns
    for (int i = threadIdx.x; i < HDIM * KPAD; i += blockDim.x) {
        const int h = i >> 4;                  // i / 16
        const int k = i & 15;                  // i % 16
        sW[i] = (k < KDIM) ? W[h * KDIM + k] : 0.0f;
    }
    __syncthreads();

    const int wave = threadIdx.x >> 5;
    const int lane = threadIdx.x & 31;
    const int tile = blockIdx.x * 8 + wave;
    const long row0 = (long)tile * 16;
    if (row0 >= nrows) return;

    const int half = lane >> 4;                // 0: K=h..h+1, 1: K=h+2..h+3
    const int mrow = lane & 15;                // M index (A) and N index (B/C)
    const float* xrow = x + (row0 + mrow) * HDIM;
    const float* sWlane = sW + 2 * half * KPAD + mrow;  // row k0 = h + 2*half

    v8f c = {};
    for (int h = 0; h < HDIM; h += 4) {
        const int k0 = h + 2 * half;
        // A layout (32-bit A 16x4): VGPR0 = K = {0|2}, VGPR1 = K = {1|3}
        v2f a = *(const v2f*)(xrow + k0);      // 8-byte aligned (k0 even)
        // B layout mirrors A's K ordering across lane halves; N = mrow
        v2f b;
        b.x = sWlane[h * KPAD];                // K = k0
        b.y = sWlane[h * KPAD + KPAD];         // K = k0 + 1
        c = __builtin_amdgcn_wmma_f32_16x16x4_f32(
                /*neg_a=*/false, a, /*neg_b=*/false, b,
                /*c_mod=*/(short)0, c, /*reuse_a=*/false, /*reuse_b=*/false);
    }

    // C layout: VGPR v -> M = v (lanes 0-15) or v+8 (lanes 16-31); N = mrow
    if (mrow < KDIM) {
        const float bb = bias[mrow];
#pragma unroll
        for (int v = 0; v < 8; ++v) {
            const long r = row0 + v + 8 * half;
            feats[r * KDIM + mrow] = c[v] + bb;
        }
    }
}

// =====================================================================
// Kernel 2: Viterbi decode per batch. One wave (32 lanes) per sequence.
// Lane j (< 6) owns forward var fv[j]; cross-state values broadcast with
// wave32 shuffles. Backpointers kept as bytes in LDS; lane 0 backtraces.
// Tie rule: strict '>' keeps the FIRST maximum, matching jnp.argmax.
// =====================================================================
__global__ __launch_bounds__(32) void
bert_crf_viterbi(const float* __restrict__ feats,
                 const float* __restrict__ trans,
                 float* __restrict__ score_out,   // [B]
                 float* __restrict__ path_out) {  // [B*T], tags as float
    __shared__ float sfeat[TDIM * KDIM];           // 12 KB
    __shared__ unsigned char sbp[TDIM * KDIM];     // 3 KB backpointers

    const int bi = blockIdx.x;
    const int lane = threadIdx.x;

    for (int i = lane; i < TDIM * KDIM; i += 32)
        sfeat[i] = feats[(long)bi * TDIM * KDIM + i];
    __syncthreads();

    // trans[next=lane][prev=p]; lanes >= 6 hold dummies, never read back
    float trow[KDIM];
#pragma unroll
    for (int p = 0; p < KDIM; ++p)
        trow[p] = (lane < KDIM) ? trans[lane * KDIM + p] : NEGV;
    const float tstop = (lane < KDIM) ? trans[STOP_IDX * KDIM + lane] : NEGV;

    float fv = (lane == START_IDX) ? 0.0f : NEGV;

    for (int t = 0; t < TDIM; ++t) {
        float best = -3.4e38f;
        int arg = 0;
#pragma unroll
        for (int p = 0; p < KDIM; ++p) {
            const float fvp = __shfl(fv, p, 32);
            const float v = fvp + trow[p];
            if (v > best) { best = v; arg = p; }
        }
        const float nf = best + ((lane < KDIM) ? sfeat[t * KDIM + lane] : 0.0f);
        if (lane < KDIM) sbp[t * KDIM + lane] = (unsigned char)arg;
        fv = (lane < KDIM) ? nf : NEGV;
    }

    // terminal = fv + transitions[STOP_IDX, :]; argmax over states
    const float term = fv + tstop;
    float bestv = -3.4e38f;
    int bestj = 0;
#pragma unroll
    for (int j = 0; j < KDIM; ++j) {
        const float v = __shfl(term, j, 32);
        if (v > bestv) { bestv = v; bestj = j; }
    }
    __syncthreads();

    if (lane == 0) {
        score_out[bi] = bestv;
        int c = bestj;
        path_out[(long)bi * TDIM + (TDIM - 1)] = (float)c;
        // reverse scan: path[t-1] = bptrs[t][path[t]]; bptrs[0] discarded
        for (int t = TDIM - 1; t >= 1; --t) {
            c = sbp[t * KDIM + c];
            path_out[(long)bi * TDIM + (t - 1)] = (float)c;
        }
    }
}

// =====================================================================
extern "C" void kernel_launch(void* const* d_in, const int* in_sizes, int n_in,
                              void* d_out, int out_size, void* d_ws, size_t ws_size,
                              hipStream_t stream) {
    const float* x     = (const float*)d_in[0];   // [B,T,H]
    const float* W     = (const float*)d_in[1];   // [H,K]
    const float* bias  = (const float*)d_in[2];   // [K]
    const float* trans = (const float*)d_in[3];   // [K,K]

    float* feats = (float*)d_ws;                  // [B*T, K] = 786432 bytes
    float* out   = (float*)d_out;                 // [B] scores ++ [B*T] path

    const int nrows = BDIM * TDIM;                // 32768
    const int tiles = nrows / 16;                 // 2048 waves
    const int blocks = tiles / 8;                 // 256 blocks of 8 waves

    bert_crf_gemm_feats<<<blocks, 256, 0, stream>>>(x, W, bias, feats, nrows);
    bert_crf_viterbi<<<BDIM, 32, 0, stream>>>(feats, trans, out, out + BDIM);
}